// RoIAlign_12764642803794
// MI455X (gfx1250) — compile-verified
//
#include <hip/hip_runtime.h>
#include <hip/hip_bf16.h>

// RoIAlign on gfx1250 (MI455X).
// features: (2, 256, 200, 200) fp32 NCHW ; rois: (N, 5) fp32 ; out: (N, 256, 7, 7) fp32
// Memory-gather bound (~0.5 FLOP/byte): optimize for coalescing + latency hiding,
// not matrix ops (no shared-B structure -> WMMA inapplicable without inflating traffic).

#define RA_C   256
#define RA_H   200
#define RA_W   200
#define RA_PH  7
#define RA_PW  7
#define RA_SCALE 0.25f
#define RA_CG  4                 // channel groups per (roi,bin)
#define RA_CPG (RA_C / RA_CG)    // 64 channels per thread
#define RA_PLANE (RA_H * RA_W)   // 40000 floats per channel plane

__global__ __launch_bounds__(256) void roi_align_gfx1250_kernel(
    const float* __restrict__ features,
    const float* __restrict__ rois,
    float* __restrict__ out,
    int N) {
  const int idx = blockIdx.x * blockDim.x + threadIdx.x;
  const int total = N * (RA_PH * RA_PW) * RA_CG;   // bin fastest -> lanes share a plane
  if (idx >= total) return;

  const unsigned bin = (unsigned)idx % 49u;        // bin = ph*7 + pw
  const unsigned rest = (unsigned)idx / 49u;
  const unsigned cg = rest & (RA_CG - 1u);
  const unsigned n  = rest >> 2;

  const int ph = (int)(bin / 7u);
  const int pw = (int)(bin % 7u);

  // ---- per-roi geometry (scalar-ish, uniform across the 49-bin lane group) ----
  const float* roi = rois + (size_t)n * 5;
  const int   b   = (int)roi[0];
  const float x1  = roi[1] * RA_SCALE;
  const float y1  = roi[2] * RA_SCALE;
  const float x2  = roi[3] * RA_SCALE;
  const float y2  = roi[4] * RA_SCALE;
  const float bin_w = fmaxf(x2 - x1, 1.0f) * (1.0f / RA_PW);
  const float bin_h = fmaxf(y2 - y1, 1.0f) * (1.0f / RA_PH);

  // ---- per-bin sample geometry: 4 samples (2x2), computed once, reused 64x ----
  int   soff[4];      // y0*W + x0 within a channel plane
  float wgt[4][4];    // bilinear weights * valid * 0.25 (sample averaging folded in)
#pragma unroll
  for (int sy = 0; sy < 2; ++sy) {
#pragma unroll
    for (int sx = 0; sx < 2; ++sx) {
      const float y = y1 + ((float)ph + ((float)sy + 0.5f) * 0.5f) * bin_h;
      const float x = x1 + ((float)pw + ((float)sx + 0.5f) * 0.5f) * bin_w;
      const bool valid = (y >= -1.0f) & (y <= (float)RA_H) &
                         (x >= -1.0f) & (x <= (float)RA_W);
      const float yc  = fminf(fmaxf(y, 0.0f), (float)(RA_H - 1));
      const float xc  = fminf(fmaxf(x, 0.0f), (float)(RA_W - 1));
      const float y0f = fminf(floorf(yc), (float)(RA_H - 2));
      const float x0f = fminf(floorf(xc), (float)(RA_W - 2));
      const int   y0  = (int)y0f;
      const int   x0  = (int)x0f;
      const float ly  = yc - y0f, lx = xc - x0f;
      const float hy  = 1.0f - ly, hx = 1.0f - lx;
      const float v   = valid ? 0.25f : 0.0f;
      const int   s   = sy * 2 + sx;
      soff[s]   = y0 * RA_W + x0;      // clamped in-range even when !valid
      wgt[s][0] = hy * hx * v;
      wgt[s][1] = hy * lx * v;
      wgt[s][2] = ly * hx * v;
      wgt[s][3] = ly * lx * v;
    }
  }

  // ---- channel loop: 64 channels, 16 independent gathers per iteration ----
  const float* fbase = features + ((size_t)b * RA_C + (size_t)cg * RA_CPG) * RA_PLANE;
  float*       obase = out + ((size_t)n * RA_C + (size_t)cg * RA_CPG) * 49 + bin;

#pragma unroll 2
  for (int i = 0; i < RA_CPG; ++i) {
    const float* p = fbase + (size_t)i * RA_PLANE;

    // gfx1250 global_prefetch_b8: prime L2 two channel-planes (320 KB) ahead.
    if (i + 2 < RA_CPG) {
      const float* pn = p + 2 * RA_PLANE;
      __builtin_prefetch(pn + soff[0], 0, 1);
      __builtin_prefetch(pn + soff[3] + RA_W, 0, 1);
    }

    float acc = 0.0f;
#pragma unroll
    for (int s = 0; s < 4; ++s) {
      const float* q = p + soff[s];
      const float f00 = q[0];
      const float f01 = q[1];
      const float f10 = q[RA_W];
      const float f11 = q[RA_W + 1];
      acc = fmaf(wgt[s][0], f00, acc);
      acc = fmaf(wgt[s][1], f01, acc);
      acc = fmaf(wgt[s][2], f10, acc);
      acc = fmaf(wgt[s][3], f11, acc);
    }
    obase[(size_t)i * 49] = acc;   // lane-adjacent bins -> coalesced stores
  }
}

extern "C" void kernel_launch(void* const* d_in, const int* in_sizes, int n_in,
                              void* d_out, int out_size, void* d_ws, size_t ws_size,
                              hipStream_t stream) {
  const float* features = (const float*)d_in[0];   // (2,256,200,200) fp32
  const float* rois     = (const float*)d_in[1];   // (N,5) fp32
  float*       out      = (float*)d_out;           // (N,256,7,7) fp32

  const int N = in_sizes[1] / 5;
  const int total = N * (RA_PH * RA_PW) * RA_CG;
  const int block = 256;
  const int grid  = (total + block - 1) / block;

  roi_align_gfx1250_kernel<<<grid, block, 0, stream>>>(features, rois, out, N);
}